// GXformer_7095285973746
// MI455X (gfx1250) — compile-verified
//
#include <hip/hip_runtime.h>

typedef __attribute__((ext_vector_type(2))) float v2f;
typedef __attribute__((ext_vector_type(8))) float v8f;

#define NB   4
#define NPT  4096
#define CIN  32
#define CN   64
#define MM   64

// ---------------------------------------------------------------------------
// Stage 1: k = E @ W_ks^T, v = E @ W_vs^T via V_WMMA_F32_16X16X4_F32.
// E is [B*N, 32] row-major, W is [64, 32] row-major (so B-matrix = W^T [32,64]).
// One wave computes one 16x16 output tile; 8 WMMA steps over K=32.
// A-frag (16x4 f32): lane<16 -> row=lane, (K=kk,kk+1); lane>=16 -> row=lane-16,
// (K=kk+2,kk+3) => aligned float2 load at column kk + 2*(lane>>4).
// B-frag (4x16 f32): mirrored: lane<16 -> col=lane, (K=kk,kk+1); lane>=16 ->
// col=lane-16, (K=kk+2,kk+3). C/D: VGPR g = rows g / g+8, col = lane&15.
// ---------------------------------------------------------------------------
__global__ void __launch_bounds__(256) proj_kernel(
    const float* __restrict__ E,
    const float* __restrict__ Wk,
    const float* __restrict__ Wv,
    float* __restrict__ Kout,
    float* __restrict__ Vout)
{
  const int lane  = threadIdx.x & 31;
  const int gwave = blockIdx.x * (blockDim.x >> 5) + (threadIdx.x >> 5);
  // 1024 row-tiles x 4 col-tiles x 2 matrices = 8192 waves
  const int rowTile = gwave & 1023;
  const int ct      = (gwave >> 10) & 3;
  const int mat     = (gwave >> 12) & 1;

  const float* W   = mat ? Wv : Wk;
  float*       Out = mat ? Vout : Kout;

  const int row0 = rowTile * 16;
  const int half = lane >> 4;   // 0 or 1
  const int l15  = lane & 15;

  const float* Abase = E + (size_t)(row0 + l15) * CIN + 2 * half;
  const float* Bbase = W + (size_t)(ct * 16 + l15) * CIN + 2 * half;

  v8f acc = {};
#pragma unroll
  for (int kk = 0; kk < CIN; kk += 4) {
    v2f a = *(const v2f*)(Abase + kk);
    v2f b = *(const v2f*)(Bbase + kk);
    acc = __builtin_amdgcn_wmma_f32_16x16x4_f32(
        /*neg_a=*/false, a, /*neg_b=*/false, b,
        /*c_mod=*/(short)0, acc, /*reuse_a=*/false, /*reuse_b=*/false);
  }

  float* obase = Out + (size_t)(row0 + 8 * half) * CN + ct * 16 + l15;
#pragma unroll
  for (int g = 0; g < 8; ++g) obase[(size_t)g * CN] = acc[g];
}

// ---------------------------------------------------------------------------
// Stage 2: farthest point sampling. One workgroup per (tensor, batch) run:
// run = t*4 + b, t in {0:events(C=32), 1:k(C=64), 2:v(C=64)}.
// Deterministic argmax with first-index tie-break (matches jnp.argmax).
// Emits the 64 selected indices sorted ascending (masked_select order).
// ---------------------------------------------------------------------------
__global__ void __launch_bounds__(256) fps_kernel(
    const float* __restrict__ events,
    const float* __restrict__ kws,
    const float* __restrict__ vws,
    int* __restrict__ idx_out)
{
  __shared__ float s_dist[NPT];
  __shared__ float s_cent[CN];
  __shared__ float s_rv[256];
  __shared__ int   s_ri[256];
  __shared__ float s_part[256];
  __shared__ int   s_sel[MM];
  __shared__ int   s_far;

  const int run = blockIdx.x;
  const int t   = run >> 2;
  const int b   = run & 3;
  const int C   = (t == 0) ? CIN : CN;
  const float* src = (t == 0) ? events + (size_t)b * NPT * CIN
                   : (t == 1) ? kws    + (size_t)b * NPT * CN
                              : vws    + (size_t)b * NPT * CN;

  const int tid = threadIdx.x;

  // ---- barycenter (deterministic two-level sum) ----
  {
    const int jper = 256 / C;          // 8 (C=32) or 4 (C=64)
    const int c    = tid % C;
    const int j    = tid / C;
    const int rows = NPT / jper;       // 512 or 1024
    float a = 0.f;
    for (int r = j * rows; r < (j + 1) * rows; ++r)
      a += src[(size_t)r * C + c];
    s_part[tid] = a;
    __syncthreads();
    if (tid < C) {
      float s = 0.f;
      for (int j2 = 0; j2 < jper; ++j2) s += s_part[j2 * C + tid];
      s_cent[tid] = s * (1.0f / (float)NPT);
    }
    __syncthreads();
  }

  // ---- d0 from barycenter, init dist = 1e10, initial farthest ----
  {
    float bmax = -1.0f; int bidx = 0;
    for (int r = tid * 16; r < tid * 16 + 16; ++r) {
      const float* row = src + (size_t)r * C;
      float d = 0.f;
      for (int cc = 0; cc < C; ++cc) { float x = row[cc] - s_cent[cc]; d += x * x; }
      s_dist[r] = 1e10f;
      if (d > bmax) { bmax = d; bidx = r; }
    }
    s_rv[tid] = bmax; s_ri[tid] = bidx;
    __syncthreads();
    if (tid < 32) {
      float m = s_rv[tid * 8]; int mi = s_ri[tid * 8];
      for (int u = 1; u < 8; ++u) {
        float v = s_rv[tid * 8 + u];
        if (v > m) { m = v; mi = s_ri[tid * 8 + u]; }
      }
      s_rv[tid] = m; s_ri[tid] = mi;
    }
    __syncthreads();
    if (tid == 0) {
      float m = s_rv[0]; int mi = s_ri[0];
      for (int u = 1; u < 32; ++u)
        if (s_rv[u] > m) { m = s_rv[u]; mi = s_ri[u]; }
      s_far = mi;
    }
    __syncthreads();
  }

  // ---- 64 sequential FPS rounds ----
  for (int it = 0; it < MM; ++it) {
    const int far = s_far;
    if (tid == 0) s_sel[it] = far;
    if (tid < C)  s_cent[tid] = src[(size_t)far * C + tid];
    __syncthreads();

    float bmax = -1.0f; int bidx = 0;
    for (int r = tid * 16; r < tid * 16 + 16; ++r) {
      const float* row = src + (size_t)r * C;
      float d = 0.f;
      for (int cc = 0; cc < C; ++cc) { float x = row[cc] - s_cent[cc]; d += x * x; }
      float dd = fminf(s_dist[r], d);
      s_dist[r] = dd;
      if (dd > bmax) { bmax = dd; bidx = r; }
    }
    s_rv[tid] = bmax; s_ri[tid] = bidx;
    __syncthreads();
    if (tid < 32) {
      float m = s_rv[tid * 8]; int mi = s_ri[tid * 8];
      for (int u = 1; u < 8; ++u) {
        float v = s_rv[tid * 8 + u];
        if (v > m) { m = v; mi = s_ri[tid * 8 + u]; }
      }
      s_rv[tid] = m; s_ri[tid] = mi;
    }
    __syncthreads();
    if (tid == 0) {
      float m = s_rv[0]; int mi = s_ri[0];
      for (int u = 1; u < 32; ++u)
        if (s_rv[u] > m) { m = s_rv[u]; mi = s_ri[u]; }
      s_far = mi;
    }
    __syncthreads();
  }

  // ---- emit selected indices in ascending (original) order via rank sort ----
  if (tid < MM) {
    int v = s_sel[tid];
    int rank = 0;
    for (int j = 0; j < MM; ++j) rank += (s_sel[j] < v) ? 1 : 0;
    idx_out[run * MM + rank] = v;
  }
}

// ---------------------------------------------------------------------------
// Stage 3: per-batch tail. Gathers + pe_m + beta + softmax + t + u  (u = [C]).
// ---------------------------------------------------------------------------
__global__ void __launch_bounds__(256) tail_kernel(
    const float* __restrict__ events,
    const float* __restrict__ kws,
    const float* __restrict__ vws,
    const int*   __restrict__ idx_sel,
    const float* __restrict__ W_pe,   // [64,32]
    const float* __restrict__ W_sa1,  // [1,64]
    const float* __restrict__ W_sa2,  // [32,64]
    float* __restrict__ u_buf)        // [B,32]
{
  __shared__ float s_evm[MM * CIN];   // 2048
  __shared__ float s_km [MM * CN];    // 4096
  __shared__ float s_vm [MM * CN];    // 4096
  __shared__ float s_pem[MM * CN];    // 4096
  __shared__ float s_part[256];
  __shared__ float s_sumE[CIN];
  __shared__ float s_P[CN];
  __shared__ float s_beta[MM];
  __shared__ float s_sv[MM];
  __shared__ float s_t[CN];
  __shared__ float s_red[2];

  const int b   = blockIdx.x;
  const int tid = threadIdx.x;
  const float* E = events + (size_t)b * NPT * CIN;
  const float* K = kws    + (size_t)b * NPT * CN;
  const float* V = vws    + (size_t)b * NPT * CN;
  const int* idxE = idx_sel + (0 * NB + b) * MM;
  const int* idxK = idx_sel + (1 * NB + b) * MM;
  const int* idxV = idx_sel + (2 * NB + b) * MM;

  // sumE[c] = sum_n E[n,c]   (deterministic two-level)
  {
    const int c = tid & 31;
    const int j = tid >> 5;                 // 8 groups of 512 rows
    float a = 0.f;
    for (int r = j * 512; r < (j + 1) * 512; ++r)
      a += E[(size_t)r * CIN + c];
    s_part[tid] = a;
    __syncthreads();
    if (tid < CIN) {
      float s = 0.f;
      for (int j2 = 0; j2 < 8; ++j2) s += s_part[j2 * 32 + tid];
      s_sumE[tid] = s;
    }
  }

  // gathers
  for (int i = tid; i < MM * CIN; i += 256) {
    int m = i >> 5, c = i & 31;
    s_evm[i] = E[(size_t)idxE[m] * CIN + c];
  }
  for (int i = tid; i < MM * CN; i += 256) {
    int m = i >> 6, c = i & 63;
    s_km[i] = K[(size_t)idxK[m] * CN + c];
    s_vm[i] = V[(size_t)idxV[m] * CN + c];
  }
  __syncthreads();

  // pe_m[m, cn] = sum_c evm[m,c] * W_pe[cn, c]
  for (int i = tid; i < MM * CN; i += 256) {
    int m = i >> 6, cn = i & 63;
    float a = 0.f;
    for (int c = 0; c < CIN; ++c) a += s_evm[m * CIN + c] * W_pe[cn * CIN + c];
    s_pem[i] = a;
  }
  __syncthreads();

  // P[cn] = sum_c sumE[c] * W_pe[cn, c] ;  beta[m] = (k_m + pe_m)[m,:] . w_sa1
  if (tid < CN) {
    float a = 0.f;
    for (int c = 0; c < CIN; ++c) a += s_sumE[c] * W_pe[tid * CIN + c];
    s_P[tid] = a;
    float bb = 0.f;
    for (int cn = 0; cn < CN; ++cn)
      bb += (s_km[tid * CN + cn] + s_pem[tid * CN + cn]) * W_sa1[cn];
    s_beta[tid] = bb;
  }
  __syncthreads();

  // scores[m] = softmax_m(-beta[m])  (n-independent!)
  if (tid == 0) {
    float mx = -s_beta[0];
    for (int m = 1; m < MM; ++m) mx = fmaxf(mx, -s_beta[m]);
    s_red[0] = mx;
  }
  __syncthreads();
  if (tid < MM) s_sv[tid] = expf(-s_beta[tid] - s_red[0]);
  __syncthreads();
  if (tid == 0) {
    float sm = 0.f;
    for (int m = 0; m < MM; ++m) sm += s_sv[m];
    s_red[1] = sm;
  }
  __syncthreads();

  // t[cn] = P[cn] + N * sum_m s_norm[m] * (v_m - pe_m)[m, cn]
  if (tid < CN) {
    const float inv = 1.0f / s_red[1];
    float a = 0.f;
    for (int m = 0; m < MM; ++m)
      a += s_sv[m] * (s_vm[m * CN + tid] - s_pem[m * CN + tid]);
    s_t[tid] = s_P[tid] + (float)NPT * a * inv;
  }
  __syncthreads();

  // u[c] = sum_cn t[cn] * W_sa2[c, cn]
  if (tid < CIN) {
    float a = 0.f;
    for (int cn = 0; cn < CN; ++cn) a += s_t[cn] * W_sa2[tid * CN + cn];
    u_buf[b * CIN + tid] = a;
  }
}

// ---------------------------------------------------------------------------
// Stage 4: broadcast u over all rows: out[b, n, c] = u[b, c]
// ---------------------------------------------------------------------------
__global__ void __launch_bounds__(256) bcast_kernel(
    const float* __restrict__ u_buf, float* __restrict__ out)
{
  const int i  = blockIdx.x * 256 + threadIdx.x;  // 4*4096*32 = 524288 total
  const int c  = i & 31;
  const int bn = i >> 5;
  const int b  = bn >> 12;                        // / 4096
  out[i] = u_buf[b * CIN + c];
}

// ---------------------------------------------------------------------------
extern "C" void kernel_launch(void* const* d_in, const int* in_sizes, int n_in,
                              void* d_out, int out_size, void* d_ws, size_t ws_size,
                              hipStream_t stream)
{
  const float* events = (const float*)d_in[0];
  // d_in[1] = W_qs: unused (cancels in softmax)
  const float* W_ks  = (const float*)d_in[2];
  const float* W_vs  = (const float*)d_in[3];
  const float* W_pe  = (const float*)d_in[4];
  const float* W_sa1 = (const float*)d_in[5];
  const float* W_sa2 = (const float*)d_in[6];

  float* kws  = (float*)d_ws;                       // [B*N*CN] = 1M floats
  float* vws  = kws + (size_t)NB * NPT * CN;        // 1M floats
  int*   idxs = (int*)(vws + (size_t)NB * NPT * CN);// 12*64 ints
  float* ubuf = (float*)(idxs + 12 * MM);           // B*32 floats

  proj_kernel <<<1024, 256, 0, stream>>>(events, W_ks, W_vs, kws, vws);
  fps_kernel  <<<12,   256, 0, stream>>>(events, kws, vws, idxs);
  tail_kernel <<<NB,   256, 0, stream>>>(events, kws, vws, idxs, W_pe, W_sa1, W_sa2, ubuf);
  bcast_kernel<<<(NB * NPT * CIN) / 256, 256, 0, stream>>>(ubuf, (float*)d_out);
}